// SpatialTransformer_block_29781303230699
// MI455X (gfx1250) — compile-verified
//
#include <hip/hip_runtime.h>
#include <stdint.h>

// VoxelMorph SpatialTransformer: trilinear warp, zero padding.
// src  : [B=2, C=2, D=160, H=192, W=160] fp32
// flow : [B=2, 3,   D,     H,     W    ] fp32  (z, y, x displacement)
// out  : [B=2, C=2, D,     H,     W    ] fp32
//
// MI455X strategy (bandwidth-bound gather, no matrix structure -> no WMMA):
//  - src (78.6 MB) fits the 192 MB L2 -> RT loads keep it resident
//  - flow/out streamed non-temporally (th:TH_*_NT) to protect L2
//  - x-corner pairs fetched with one 4B-aligned b64 load -> half the requests
//  - gathers use uniform-base + unsigned 32-bit offsets (SADDR/GVS form)
//  - edge remap of the pair weights is a single conditional swap (the
//    zero-padding masks make the other cases collapse to 0 automatically)

typedef float v4f __attribute__((ext_vector_type(4)));

struct __attribute__((packed, aligned(4))) fpair { float lo, hi; };

#define DD 160
#define HH 192
#define WW 160
#define NSP (DD * HH * WW)   // 4,915,200 spatial points per batch

__global__ __launch_bounds__(256)
void warp3d_trilinear(const float* __restrict__ src,
                      const float* __restrict__ flow,
                      float* __restrict__ out) {
    const int b    = blockIdx.y;                       // batch
    const int pack = blockIdx.x * 256 + threadIdx.x;   // 4-voxel pack id
    const int s0   = pack * 4;                         // flat spatial index
    if (s0 >= NSP) return;

    // 4 voxels share (z,y); x = x0..x0+3 (W divisible by 4, no row crossing)
    const int x0 = s0 % WW;
    const int t  = s0 / WW;
    const int yi = t % HH;
    const int zi = t / HH;

    // streaming flow loads (non-temporal)
    const float* __restrict__ fb = flow + (size_t)b * 3 * NSP;
    const v4f fz = __builtin_nontemporal_load((const v4f*)(fb + s0));
    const v4f fy = __builtin_nontemporal_load((const v4f*)(fb + NSP + s0));
    const v4f fx = __builtin_nontemporal_load((const v4f*)(fb + 2 * NSP + s0));

    const float* __restrict__ c0 = src + (size_t)b * 2 * NSP;  // channel 0
    const float* __restrict__ c1 = c0 + NSP;                   // channel 1

    v4f o0, o1;

#pragma unroll
    for (int i = 0; i < 4; ++i) {
        const float z = (float)zi + fz[i];
        const float y = (float)yi + fy[i];
        const float x = (float)(x0 + i) + fx[i];

        const float zf = floorf(z), yf = floorf(y), xf = floorf(x);
        const int   z0 = (int)zf,  y0 = (int)yf,  xq = (int)xf;
        const float az = z - zf,   ay = y - yf,   ax = x - xf;

        // per-axis weights with zero-padding masks folded in
        const float wz0 = ((unsigned)z0       < (unsigned)DD) ? (1.0f - az) : 0.0f;
        const float wz1 = ((unsigned)(z0 + 1) < (unsigned)DD) ? az          : 0.0f;
        const float wy0 = ((unsigned)y0       < (unsigned)HH) ? (1.0f - ay) : 0.0f;
        const float wy1 = ((unsigned)(y0 + 1) < (unsigned)HH) ? ay          : 0.0f;
        const float wxl = ((unsigned)xq       < (unsigned)WW) ? (1.0f - ax) : 0.0f;
        const float wxh = ((unsigned)(xq + 1) < (unsigned)WW) ? ax          : 0.0f;

        // one b64 pair load covers both x corners, based at xc=clamp(xq,0,W-2).
        // Off-interior (xq<0 or xq>W-2) the pair weights just swap:
        //   xq = -1 : corner x=0   is the pair's lo element, weight wxh (wxl==0)
        //   xq = W-1: corner x=W-1 is the pair's hi element, weight wxl (wxh==0)
        //   fully OOB: both masked weights are already 0, swap is harmless
        const int  xc   = min(max(xq, 0), WW - 2);
        const bool swap = ((unsigned)xq >= (unsigned)(WW - 1));  // xq<0 || xq>W-2
        const float a0 = swap ? wxh : wxl;
        const float a1 = swap ? wxl : wxh;

        const int zc0 = min(max(z0,     0), DD - 1);
        const int zc1 = min(max(z0 + 1, 0), DD - 1);
        const int yc0 = min(max(y0,     0), HH - 1);
        const int yc1 = min(max(y0 + 1, 0), HH - 1);

        // unsigned 32-bit row offsets, shared by both channels (SADDR form)
        const uint32_t r00 = (uint32_t)((zc0 * HH + yc0) * WW + xc);
        const uint32_t r01 = (uint32_t)((zc0 * HH + yc1) * WW + xc);
        const uint32_t r10 = (uint32_t)((zc1 * HH + yc0) * WW + xc);
        const uint32_t r11 = (uint32_t)((zc1 * HH + yc1) * WW + xc);

        const float w00 = wz0 * wy0;
        const float w01 = wz0 * wy1;
        const float w10 = wz1 * wy0;
        const float w11 = wz1 * wy1;

        // issue all 8 pair-gathers up front for max memory-level parallelism
        const fpair g000 = *(const fpair*)(c0 + r00);
        const fpair g001 = *(const fpair*)(c0 + r01);
        const fpair g010 = *(const fpair*)(c0 + r10);
        const fpair g011 = *(const fpair*)(c0 + r11);
        const fpair g100 = *(const fpair*)(c1 + r00);
        const fpair g101 = *(const fpair*)(c1 + r01);
        const fpair g110 = *(const fpair*)(c1 + r10);
        const fpair g111 = *(const fpair*)(c1 + r11);

        float acc0, acc1;
        acc0 = w00 * fmaf(a1, g000.hi, a0 * g000.lo);
        acc0 = fmaf(w01, fmaf(a1, g001.hi, a0 * g001.lo), acc0);
        acc0 = fmaf(w10, fmaf(a1, g010.hi, a0 * g010.lo), acc0);
        acc0 = fmaf(w11, fmaf(a1, g011.hi, a0 * g011.lo), acc0);
        acc1 = w00 * fmaf(a1, g100.hi, a0 * g100.lo);
        acc1 = fmaf(w01, fmaf(a1, g101.hi, a0 * g101.lo), acc1);
        acc1 = fmaf(w10, fmaf(a1, g110.hi, a0 * g110.lo), acc1);
        acc1 = fmaf(w11, fmaf(a1, g111.hi, a0 * g111.lo), acc1);

        o0[i] = acc0;
        o1[i] = acc1;
    }

    // streaming output stores (non-temporal)
    float* __restrict__ ob = out + (size_t)b * 2 * NSP;
    __builtin_nontemporal_store(o0, (v4f*)(ob + s0));
    __builtin_nontemporal_store(o1, (v4f*)(ob + NSP + s0));
}

extern "C" void kernel_launch(void* const* d_in, const int* in_sizes, int n_in,
                              void* d_out, int out_size, void* d_ws, size_t ws_size,
                              hipStream_t stream) {
    const float* src  = (const float*)d_in[0];   // 2*2*160*192*160 floats
    const float* flow = (const float*)d_in[1];   // 2*3*160*192*160 floats
    float*       out  = (float*)d_out;           // 2*2*160*192*160 floats

    // NSP/4 packs per batch = 1,228,800 -> 4800 blocks of 256 threads, exact.
    dim3 grid((NSP / 4 + 255) / 256, 2, 1);
    warp3d_trilinear<<<grid, 256, 0, stream>>>(src, flow, out);
}